// KMeansCrossAttention_61349312856196
// MI455X (gfx1250) — compile-verified
//
#include <hip/hip_runtime.h>
#include <hip/hip_bf16.h>
#include <math.h>

// Problem constants from the reference (fixed by setup_inputs()).
#define B_ 4
#define N_ 1024
#define M_ 8192
#define D_ 256
#define MT 128          // kv tokens per workgroup tile
#define NT 16           // query rows per chunk (WMMA tile height)
#define THREADS 256     // 8 waves of 32; also == D_ for the scatter phase
#define NCHUNK (N_ / NT)

typedef __attribute__((ext_vector_type(2))) float v2f;
typedef __attribute__((ext_vector_type(8))) float v8f;
typedef __attribute__((ext_vector_type(4))) int   v4i;

#define NEG_INF_F (-__builtin_inff())

#define AS1 __attribute__((address_space(1)))
#define AS3 __attribute__((address_space(3)))

// Async global->LDS path (CDNA5 GLOBAL_LOAD_ASYNC_TO_LDS_*, ASYNCcnt).
#if defined(__AMDGCN__) && \
    __has_builtin(__builtin_amdgcn_global_load_async_to_lds_b128) && \
    __has_builtin(__builtin_amdgcn_global_load_async_to_lds_b8) && \
    __has_builtin(__builtin_amdgcn_s_wait_asynccnt)
#define USE_ASYNC_LDS 1
#else
#define USE_ASYNC_LDS 0
#endif

// -------------------------------------------------------------------------
// Zero-fill helper (harness poisons d_out / d_ws with 0xAA).
// -------------------------------------------------------------------------
__global__ void kca_zero_kernel(float* __restrict__ p, int n) {
    int i = blockIdx.x * blockDim.x + threadIdx.x;
    if (i < n) p[i] = 0.0f;
}

// -------------------------------------------------------------------------
// Main kernel: per kv-token argmax over queries via f32 WMMA, then scatter.
// Grid: (M_/MT, B_), Block: 256 threads (8 waves).
// Each wave owns 16 kv columns; its K^T fragments live in 128 VGPRs.
// Q chunks are double-buffered in LDS, filled by async global->LDS DMA.
// -------------------------------------------------------------------------
__global__ __launch_bounds__(THREADS)
void kca_assign_kernel(const float* __restrict__ q,
                       const float* __restrict__ k,
                       const float* __restrict__ v,
                       const unsigned char* __restrict__ q_mask,   // jnp bool -> 1 byte
                       const unsigned char* __restrict__ kv_mask,  // jnp bool -> 1 byte
                       float* __restrict__ out,
                       float* __restrict__ counts)
{
    __shared__ float         Qs[2][NT * D_];    // 2 x 16 KB double buffer
    __shared__ unsigned char qm_s[2][NT];
    __shared__ int           s_idx[MT];
    __shared__ unsigned char s_valid[MT];

    const int b      = blockIdx.y;
    const int m_base = blockIdx.x * MT;
    const int tid    = threadIdx.x;
    const int lane   = tid & 31;
    const int wave   = tid >> 5;            // 0..7
    const int hi     = (lane >> 4) & 1;     // lane half selects K sub-pair
    const int l16    = lane & 15;

    // ---- Preload this wave's B-matrix fragments (K^T, 16 kv columns) -------
    // V_WMMA_F32_16X16X4_F32 B layout (4xK x 16 cols): lane = column,
    // lanes 0-15 hold K=0,1 in VGPR0,1; lanes 16-31 hold K=2,3.
    const int     mcol = m_base + wave * 16 + l16;
    const float*  krow = k + ((size_t)b * M_ + mcol) * D_;
    v2f bfrag[D_ / 4];                       // 64 x float2 = 128 VGPRs
    #pragma unroll
    for (int i = 0; i < D_ / 4; ++i) {
        const float* p = krow + i * 4 + hi * 2;
        bfrag[i].x = p[0];
        bfrag[i].y = p[1];
    }

    // ---- Stage helper: bring query chunk c into LDS buffer (c & 1) ---------
    auto stage_chunk = [&](int c) {
        const int  buf  = c & 1;
        const float* qsrc = q + ((size_t)b * N_ + c * NT) * D_;
#if USE_ASYNC_LDS
        #pragma unroll
        for (int j = 0; j < (NT * D_ / 4) / THREADS; ++j) {      // 4 x b128 / thread
            int i = tid + j * THREADS;                            // float4 index
            __builtin_amdgcn_global_load_async_to_lds_b128(
                (AS1 v4i*)(qsrc + i * 4),
                (AS3 v4i*)(&Qs[buf][i * 4]),
                0, 0);
        }
        if (tid < NT)
            __builtin_amdgcn_global_load_async_to_lds_b8(
                (AS1 char*)(q_mask + b * N_ + c * NT + tid),
                (AS3 char*)(&qm_s[buf][tid]),
                0, 0);
#else
        #pragma unroll
        for (int j = 0; j < (NT * D_ / 4) / THREADS; ++j) {
            int i = tid + j * THREADS;
            ((float4*)Qs[buf])[i] = ((const float4*)qsrc)[i];
        }
        if (tid < NT) qm_s[buf][tid] = q_mask[b * N_ + c * NT + tid];
#endif
    };

    float runbest = NEG_INF_F;   // running max over n for this lane's kv column
    int   runidx  = 0;           // running argmax (first occurrence wins)

    stage_chunk(0);              // prologue: chunk 0 in flight

    for (int c = 0; c < NCHUNK; ++c) {
        const int buf = c & 1;
        const int n0  = c * NT;

#if USE_ASYNC_LDS
        __builtin_amdgcn_s_wait_asynccnt(0);   // my portion of buf landed
#endif
        __syncthreads();                        // everyone's portion landed;
                                                // prior reads of other buf done
        if (c + 1 < NCHUNK)
            stage_chunk(c + 1);                 // overlap next load with compute

        // 16(n) x 16(m) logits tile, accumulated over D in K=4 steps.
        // A layout (16 rows x 4): lane row = lane&15; lane half selects K pair.
        v8f acc = {};
        #pragma unroll
        for (int i = 0; i < D_ / 4; ++i) {
            v2f a;
            const float* ap = &Qs[buf][l16 * D_ + i * 4 + hi * 2];
            a.x = ap[0];
            a.y = ap[1];
            acc = __builtin_amdgcn_wmma_f32_16x16x4_f32(
                      false, a, false, bfrag[i], (short)0, acc, false, false);
        }

        // Per-column max/argmax over this tile's 16 query rows.
        // C/D layout: lane holds column (lane&15); VGPR r = row r + 8*hi.
        float best  = NEG_INF_F;
        int   bestn = n0;
        #pragma unroll
        for (int r = 0; r < 8; ++r) {
            int   nl  = r + 8 * hi;
            float val = qm_s[buf][nl] ? acc[r] : NEG_INF_F;
            if (val > best) { best = val; bestn = n0 + nl; }
        }
        // Combine lane halves (rows 0-7 vs 8-15); prefer smaller n on ties.
        float ob = __shfl_xor(best, 16, 32);
        int   on = __shfl_xor(bestn, 16, 32);
        if (ob > best || (ob == best && on < bestn)) { best = ob; bestn = on; }
        // Running update: earlier chunks win ties via strict '>'.
        if (best > runbest) { runbest = best; runidx = bestn; }
    }

    // ---- Publish per-kv-token assignment ----------------------------------
    if (lane < 16) {
        int  mloc  = wave * 16 + l16;
        bool valid = (kv_mask[b * M_ + m_base + mloc] != 0) &&
                     (runbest > NEG_INF_F);
        s_idx[mloc]   = runidx;
        s_valid[mloc] = valid ? 1 : 0;
    }
    __syncthreads();

    // counts[b, idx] += 1 per valid kv token in this tile
    if (tid < MT && s_valid[tid])
        atomicAdd(&counts[b * N_ + s_idx[tid]], 1.0f);

    // out[b, idx, :] += v[b, m, :]   (tid == d since THREADS == D_)
    for (int mm = 0; mm < MT; ++mm) {
        if (s_valid[mm]) {
            float val = v[((size_t)b * M_ + m_base + mm) * D_ + tid];
            atomicAdd(&out[((size_t)b * N_ + s_idx[mm]) * D_ + tid], val);
        }
    }
}

// -------------------------------------------------------------------------
// Finalize: out /= (clip(counts, 1) + eps)
// -------------------------------------------------------------------------
__global__ void kca_finalize_kernel(float* __restrict__ out,
                                    const float* __restrict__ counts)
{
    int i = blockIdx.x * blockDim.x + threadIdx.x;   // over B*N*D
    if (i >= B_ * N_ * D_) return;
    float c = counts[i / D_];
    c = fmaxf(c, 1.0f) + 1e-6f;
    out[i] = out[i] / c;
}

// -------------------------------------------------------------------------
extern "C" void kernel_launch(void* const* d_in, const int* in_sizes, int n_in,
                              void* d_out, int out_size, void* d_ws, size_t ws_size,
                              hipStream_t stream)
{
    const float*         q   = (const float*)d_in[0];          // [B,N,D]
    const float*         k   = (const float*)d_in[1];          // [B,M,D]
    const float*         v   = (const float*)d_in[2];          // [B,M,D]
    const unsigned char* qm  = (const unsigned char*)d_in[3];  // [B,N] bool
    const unsigned char* kvm = (const unsigned char*)d_in[4];  // [B,M] bool
    float*               out = (float*)d_out;                  // [B,N,D]
    float*               cnt = (float*)d_ws;                   // [B,N] scratch (16 KB)

    (void)in_sizes; (void)n_in; (void)out_size; (void)ws_size;

    kca_zero_kernel<<<(B_ * N_ * D_ + 255) / 256, 256, 0, stream>>>(out, B_ * N_ * D_);
    kca_zero_kernel<<<(B_ * N_ + 255) / 256, 256, 0, stream>>>(cnt, B_ * N_);

    dim3 grid(M_ / MT, B_);   // (64, 4) workgroups
    kca_assign_kernel<<<grid, THREADS, 0, stream>>>(q, k, v, qm, kvm, out, cnt);

    kca_finalize_kernel<<<(B_ * N_ * D_ + 255) / 256, 256, 0, stream>>>(out, cnt);
}